// NGCF_19902878449772
// MI455X (gfx1250) — compile-verified
//
#include <hip/hip_runtime.h>

// NGCF 3-layer propagation for MI455X (gfx1250, wave32).
// f32 end-to-end: GEMM FLOPs (~7.4G total) are negligible on this chip; the
// workload is L2/HBM bound, so we keep exact f32 semantics and use
// V_WMMA_F32_16X16X4_F32 for the dense [N,64]x[64,64] GEMMs.

#define N_USER 100000
#define N_ITEM 50000
#define NN     150000      /* N_USER + N_ITEM */
#define DD     64
#define NEDGE  1200000
#define SLOPE  0.01f
#define EPSN   1e-12f

typedef __attribute__((ext_vector_type(2))) float v2f;
typedef __attribute__((ext_vector_type(8))) float v8f;

// ---------------------------------------------------------------------------
// E0 = concat(user, item); out[:, 0:64] = E0 (un-normalized per reference)
// ---------------------------------------------------------------------------
__global__ __launch_bounds__(256) void init_e0(const float* __restrict__ user,
                                               const float* __restrict__ item,
                                               float* __restrict__ Ecur,
                                               float* __restrict__ out) {
    int idx4 = blockIdx.x * blockDim.x + threadIdx.x;   // over N*64/4
    if (idx4 >= NN * (DD / 4)) return;
    int fidx = idx4 * 4;
    const float* src = (fidx < N_USER * DD) ? (user + fidx)
                                            : (item + (fidx - N_USER * DD));
    float4 v = *(const float4*)src;
    *(float4*)(Ecur + fidx) = v;
    int row = fidx >> 6;
    int col = fidx & 63;
    *(float4*)(out + (size_t)row * 256 + col) = v;
}

// ---------------------------------------------------------------------------
// front = E   (the +I term; scatter adds the H part on top)
// ---------------------------------------------------------------------------
__global__ __launch_bounds__(256) void copy_f4(const float* __restrict__ src,
                                               float* __restrict__ dst, int n4) {
    int i = blockIdx.x * blockDim.x + threadIdx.x;
    if (i >= n4) return;
    *(float4*)(dst + (size_t)i * 4) = *(const float4*)(src + (size_t)i * 4);
}

// ---------------------------------------------------------------------------
// front[row] += val * E[col]  : one thread per (edge, feature)
// ---------------------------------------------------------------------------
__global__ __launch_bounds__(256) void spmm_scatter(const int* __restrict__ erow,
                                                    const int* __restrict__ ecol,
                                                    const float* __restrict__ eval,
                                                    const float* __restrict__ E,
                                                    float* __restrict__ front) {
    int idx = blockIdx.x * blockDim.x + threadIdx.x;    // over NEDGE*64
    int e = idx >> 6;
    int d = idx & 63;
    if (e >= NEDGE) return;
    int r = erow[e];
    int c = ecol[e];
    float v = eval[e];
    atomicAdd(front + (size_t)r * DD + d, v * E[(size_t)c * DD + d]);
}

// ---------------------------------------------------------------------------
// Fused:  Enext = leaky(front @ Wf^T + bf) + leaky((E*front) @ Wb^T + bb)
// One block = one 16-row tile, 4 waves = 4 column tiles of 16.
// A fragments from LDS; B fragments (B[k][n] = W[n][k]) from global (W is
// 16KB, fully cache-resident).  Enext may alias front (block only touches its
// own rows, which are staged into LDS before any store).
// ---------------------------------------------------------------------------
__global__ __launch_bounds__(128) void gemm_fused(const float* __restrict__ E,
                                                  const float* __restrict__ F,
                                                  float* __restrict__ O,
                                                  const float* __restrict__ Wf,
                                                  const float* __restrict__ bf,
                                                  const float* __restrict__ Wb,
                                                  const float* __restrict__ bb) {
    __shared__ float sF[16 * DD];   // front tile
    __shared__ float sX2[16 * DD];  // (E * front) tile

    const int r0  = blockIdx.x * 16;
    const int tid = threadIdx.x;            // 0..127

    // stage 16x64 tiles: 1024 floats, 128 threads x 2 float4
    for (int it = 0; it < 2; ++it) {
        int off = (it * 128 + tid) * 4;
        float4 fv = *(const float4*)(F + (size_t)r0 * DD + off);
        float4 ev = *(const float4*)(E + (size_t)r0 * DD + off);
        *(float4*)(sF + off) = fv;
        float4 x2;
        x2.x = ev.x * fv.x; x2.y = ev.y * fv.y;
        x2.z = ev.z * fv.z; x2.w = ev.w * fv.w;
        *(float4*)(sX2 + off) = x2;
    }
    __syncthreads();

    const int wave = tid >> 5;              // wave32: 4 waves/block
    const int lane = tid & 31;
    const int hi   = lane >> 4;             // 0: K+0,1 | 1: K+2,3 (A/B layout)
    const int l16  = lane & 15;
    const int n    = wave * 16 + l16;       // output column (0..63)

    v8f acc1 = {};
    v8f acc2 = {};

    for (int kb = 0; kb < 16; ++kb) {       // K = 64 in steps of 4
        const int ks = kb * 4 + hi * 2;
        v2f a1 = *(const v2f*)(sF  + l16 * DD + ks);
        v2f a2 = *(const v2f*)(sX2 + l16 * DD + ks);
        v2f b1 = *(const v2f*)(Wf + (size_t)n * DD + ks);
        v2f b2 = *(const v2f*)(Wb + (size_t)n * DD + ks);
        acc1 = __builtin_amdgcn_wmma_f32_16x16x4_f32(false, a1, false, b1,
                                                     (short)0, acc1, false, false);
        acc2 = __builtin_amdgcn_wmma_f32_16x16x4_f32(false, a2, false, b2,
                                                     (short)0, acc2, false, false);
    }

    // C/D layout: lane<16 -> rows v, lane>=16 -> rows v+8 (v = VGPR index)
    const float bfn = bf[n];
    const float bbn = bb[n];
#pragma unroll
    for (int v = 0; v < 8; ++v) {
        int row = r0 + hi * 8 + v;
        float x1 = acc1[v] + bfn;  x1 = (x1 > 0.0f) ? x1 : SLOPE * x1;
        float x2 = acc2[v] + bbn;  x2 = (x2 > 0.0f) ? x2 : SLOPE * x2;
        O[(size_t)row * DD + n] = x1 + x2;
    }
}

// ---------------------------------------------------------------------------
// out[:, (layer+1)*64 : (layer+2)*64] = Enew / max(||Enew||_2, eps)
// one wave per row, butterfly reduction (wave32)
// ---------------------------------------------------------------------------
__global__ __launch_bounds__(256) void normalize_store(const float* __restrict__ Enew,
                                                       float* __restrict__ out,
                                                       int layer) {
    int row  = blockIdx.x * 8 + (threadIdx.x >> 5);
    int lane = threadIdx.x & 31;
    if (row >= NN) return;
    float2 v = *(const float2*)(Enew + (size_t)row * DD + lane * 2);
    float ss = v.x * v.x + v.y * v.y;
#pragma unroll
    for (int m = 16; m >= 1; m >>= 1) ss += __shfl_xor(ss, m, 32);
    float nrm = fmaxf(sqrtf(ss), EPSN);
    float inv = 1.0f / nrm;
    float2 o;
    o.x = v.x * inv;
    o.y = v.y * inv;
    *(float2*)(out + (size_t)row * 256 + (layer + 1) * 64 + lane * 2) = o;
}

// ---------------------------------------------------------------------------
extern "C" void kernel_launch(void* const* d_in, const int* in_sizes, int n_in,
                              void* d_out, int out_size, void* d_ws, size_t ws_size,
                              hipStream_t stream) {
    const float* user = (const float*)d_in[0];
    const float* item = (const float*)d_in[1];
    const int*   erow = (const int*)d_in[2];
    const int*   ecol = (const int*)d_in[3];
    const float* eval = (const float*)d_in[4];
    const float* Wf   = (const float*)d_in[5];
    const float* bf   = (const float*)d_in[6];
    const float* Wb   = (const float*)d_in[7];
    const float* bb   = (const float*)d_in[8];
    float* out = (float*)d_out;

    const size_t MAT = (size_t)NN * DD;     // 9.6M floats
    float* A = (float*)d_ws;                // ping
    float* B = A + MAT;                     // pong

    // layer 0 input: E0 = concat(user,item); also out[:,0:64]
    init_e0<<<(NN * (DD / 4) + 255) / 256, 256, 0, stream>>>(user, item, A, out);

    float* ecur = A;
    float* fr   = B;
    for (int i = 0; i < 3; ++i) {
        copy_f4<<<(int)((MAT / 4 + 255) / 256), 256, 0, stream>>>(ecur, fr, (int)(MAT / 4));
        spmm_scatter<<<(NEDGE * DD) / 256, 256, 0, stream>>>(erow, ecol, eval, ecur, fr);
        gemm_fused<<<NN / 16, 128, 0, stream>>>(ecur, fr, fr,
                                                Wf + (size_t)i * DD * DD, bf + (size_t)i * DD,
                                                Wb + (size_t)i * DD * DD, bb + (size_t)i * DD);
        normalize_store<<<(NN + 7) / 8, 256, 0, stream>>>(fr, out, i);
        float* t = ecur; ecur = fr; fr = t;   // E_{l+1} (un-normalized) ping-pong
    }
}